// GNN_21534966022499
// MI455X (gfx1250) — compile-verified
//
#include <hip/hip_runtime.h>

// ---------------------------------------------------------------------------
// GCN (2-layer) for MI455X / gfx1250.
//   N = 100000 nodes, E = 1,600,000 edges (+N self loops), 128 -> 128 -> 64.
//   GEMMs on V_WMMA_F32_16X16X4_F32 (fp32 matrix path, exact-precision).
//   Edge gather/scatter: one wave32 per edge, row-contiguous coalesced loads,
//   global_atomic_add_f32 scatter (L2-resident: 51MB features << 192MB L2).
// ---------------------------------------------------------------------------

typedef float v2f __attribute__((ext_vector_type(2)));
typedef float v8f __attribute__((ext_vector_type(8)));

// ---------------- utility: zero a buffer (graph-capture safe) ---------------
__global__ __launch_bounds__(256) void gcn_zero_f4(float4* p, int n4) {
    int i = blockIdx.x * 256 + threadIdx.x;
    if (i < n4) p[i] = make_float4(0.f, 0.f, 0.f, 0.f);
}

// ---------------- degree / normalization ------------------------------------
__global__ __launch_bounds__(256) void gcn_deg_init(float* deg, int N) {
    int i = blockIdx.x * 256 + threadIdx.x;
    if (i < N) deg[i] = 1.0f;                       // self-loop contribution
}

__global__ __launch_bounds__(256) void gcn_deg_edges(const long long* __restrict__ ei,
                                                     long long E, float* deg) {
    long long e = (long long)blockIdx.x * 256 + threadIdx.x;
    if (e < E) atomicAdd(&deg[ei[E + e]], 1.0f);    // dst row of edge_index
}

__global__ __launch_bounds__(256) void gcn_dinv(float* deg, int N) {
    int i = blockIdx.x * 256 + threadIdx.x;
    if (i < N) {
        float d = deg[i];
        deg[i] = (d > 0.f) ? rsqrtf(d) : 0.f;       // in place: deg -> dinv
    }
}

// ---------------- WMMA fp32 GEMM:  C[16 x NCOL] tile per block --------------
// K is always 128.  One wave per 16-column tile (NCOL/16 waves per block).
// A tile staged in LDS with +4-float row padding (bank-conflict free for the
// 16-lane row-strided ds_load_b64 pattern).  W streamed from global (L0/L2
// resident: W1 = 64KB total, identical for all 6250 blocks).
template <int NCOL>
__global__ __launch_bounds__(32 * (NCOL / 16))
void gcn_gemm_wmma(const float* __restrict__ A, const float* __restrict__ W,
                   float* __restrict__ C) {
    constexpr int K  = 128;
    constexpr int XS = K + 4;                        // padded LDS row stride
    __shared__ float sX[16 * XS];

    const int tid      = threadIdx.x;
    const int nthreads = 32 * (NCOL / 16);
    const long long rowBase = (long long)blockIdx.x * 16;   // N == 6250*16 exact

    // stage 16 x 128 A tile
    for (int i = tid * 4; i < 16 * K; i += nthreads * 4) {
        int r = i / K, c = i % K;
        float4 v = *reinterpret_cast<const float4*>(A + (rowBase + r) * K + c);
        *reinterpret_cast<float4*>(&sX[r * XS + c]) = v;
    }
    __syncthreads();

    const int lane = tid & 31;
    const int n0   = (tid >> 5) * 16;                // this wave's column tile
    const int m    = lane & 15;                      // A row for this lane
    const int kb   = (lane >> 4) * 2;                // K-pair select (0 or 2)
    const int n    = lane & 15;                      // B/D column for this lane

    const float* wp = W + n0 + n;                    // column pointer into W

    v8f acc = {0.f, 0.f, 0.f, 0.f, 0.f, 0.f, 0.f, 0.f};
#pragma unroll
    for (int k0 = 0; k0 < K; k0 += 4) {
        v2f a, b;
        a.x = sX[m * XS + k0 + kb];                  // one ds_load_b64
        a.y = sX[m * XS + k0 + kb + 1];
        b.x = wp[(k0 + kb) * NCOL];                  // global, L0-hot
        b.y = wp[(k0 + kb + 1) * NCOL];
        acc = __builtin_amdgcn_wmma_f32_16x16x4_f32(
            /*neg_a=*/false, a, /*neg_b=*/false, b,
            /*c_mod=*/(short)0, acc, /*reuse_a=*/false, /*reuse_b=*/false);
    }

    // D layout: VGPR r -> rows r (lanes 0-15) and r+8 (lanes 16-31)
    const int mHi = (lane >> 4) * 8;
#pragma unroll
    for (int r = 0; r < 8; ++r)
        C[(rowBase + mHi + r) * NCOL + n0 + n] = acc[r];
}

// ---------------- edge gather * norm -> atomic scatter-sum ------------------
// One wave per edge.  Lane l covers F/32 consecutive features: the wave's
// load is one contiguous F*4-byte segment (512B / 256B), fully coalesced.
template <int F>
__global__ __launch_bounds__(256)
void gcn_scatter(const long long* __restrict__ ei, long long E, int N,
                 const float* __restrict__ dinv,
                 const float* __restrict__ feat,
                 float* __restrict__ agg) {
    constexpr int CH = F / 32;
    const long long e = (long long)blockIdx.x * 8 + (threadIdx.x >> 5);
    if (e >= E + (long long)N) return;
    const int lane = threadIdx.x & 31;

    long long s, d;
    if (e < E) { s = ei[e]; d = ei[E + e]; }
    else       { s = e - E; d = s; }                 // self loop
    const float nrm = dinv[s] * dinv[d];

    const float* sp = feat + s * F + lane * CH;
    float*       dp = agg  + d * F + lane * CH;
    if constexpr (CH == 4) {
        float4 v = *reinterpret_cast<const float4*>(sp);
        atomicAdd(dp + 0, v.x * nrm);
        atomicAdd(dp + 1, v.y * nrm);
        atomicAdd(dp + 2, v.z * nrm);
        atomicAdd(dp + 3, v.w * nrm);
    } else {
        float2 v = *reinterpret_cast<const float2*>(sp);
        atomicAdd(dp + 0, v.x * nrm);
        atomicAdd(dp + 1, v.y * nrm);
    }
}

// ---------------- bias + ReLU (layer 1 epilogue), in place ------------------
__global__ __launch_bounds__(256) void gcn_bias_relu(float* __restrict__ h,
                                                     const float* __restrict__ b,
                                                     int n4) {
    int i = blockIdx.x * 256 + threadIdx.x;          // float4 index
    if (i >= n4) return;
    float4 v = reinterpret_cast<float4*>(h)[i];
    int c = (i * 4) & 127;
    v.x = fmaxf(v.x + b[c + 0], 0.f);
    v.y = fmaxf(v.y + b[c + 1], 0.f);
    v.z = fmaxf(v.z + b[c + 2], 0.f);
    v.w = fmaxf(v.w + b[c + 3], 0.f);
    reinterpret_cast<float4*>(h)[i] = v;
}

// ---------------- bias + log_softmax over 64 classes, one wave per node -----
__global__ __launch_bounds__(256) void gcn_bias_logsoftmax(float* __restrict__ out,
                                                           const float* __restrict__ b2,
                                                           int N) {
    int node = blockIdx.x * 8 + (threadIdx.x >> 5);
    if (node >= N) return;
    int lane = threadIdx.x & 31;
    float* row = out + (long long)node * 64;
    float2 v = *reinterpret_cast<float2*>(row + lane * 2);
    v.x += b2[lane * 2 + 0];
    v.y += b2[lane * 2 + 1];

    float mx = fmaxf(v.x, v.y);
#pragma unroll
    for (int off = 16; off > 0; off >>= 1) mx = fmaxf(mx, __shfl_xor(mx, off, 32));
    float s = __expf(v.x - mx) + __expf(v.y - mx);
#pragma unroll
    for (int off = 16; off > 0; off >>= 1) s += __shfl_xor(s, off, 32);
    float lse = __logf(s) + mx;

    row[lane * 2 + 0] = v.x - lse;
    row[lane * 2 + 1] = v.y - lse;
}

// ---------------------------------------------------------------------------
extern "C" void kernel_launch(void* const* d_in, const int* in_sizes, int n_in,
                              void* d_out, int out_size, void* d_ws, size_t ws_size,
                              hipStream_t stream) {
    const float*     x  = (const float*)d_in[0];        // [N,128]
    const long long* ei = (const long long*)d_in[1];    // [2,E] int64
    const float*     W1 = (const float*)d_in[2];        // [128,128]
    const float*     b1 = (const float*)d_in[3];        // [128]
    const float*     W2 = (const float*)d_in[4];        // [128,64]
    const float*     b2 = (const float*)d_in[5];        // [64]
    float*           out = (float*)d_out;               // [N,64]

    const int       N = in_sizes[0] / 128;              // 100000 (== 6250*16)
    const long long E = in_sizes[1] / 2;                // 1,600,000

    // workspace layout
    char*  ws   = (char*)d_ws;
    float* dinv = (float*)ws;                                        // [N]
    size_t off  = (((size_t)N * 4) + 255) & ~(size_t)255;
    float* buf1 = (float*)(ws + off);                                // [N,128]
    float* buf2 = (float*)(ws + off + (size_t)N * 128 * sizeof(float)); // [N,128]

    const long long tot    = E + (long long)N;
    const int edgeBlocks   = (int)((tot + 7) / 8);
    const int nBlocks256   = (N + 255) / 256;

    // --- degrees -> dinv ---
    gcn_deg_init<<<nBlocks256, 256, 0, stream>>>(dinv, N);
    gcn_deg_edges<<<(int)((E + 255) / 256), 256, 0, stream>>>(ei, E, dinv);
    gcn_dinv<<<nBlocks256, 256, 0, stream>>>(dinv, N);

    // --- layer 1: t1 = x @ W1 ; agg = scatter(t1) ; h = relu(agg + b1) ---
    gcn_gemm_wmma<128><<<N / 16, 256, 0, stream>>>(x, W1, buf1);
    gcn_zero_f4<<<(N * 32 + 255) / 256, 256, 0, stream>>>((float4*)buf2, N * 32);
    gcn_scatter<128><<<edgeBlocks, 256, 0, stream>>>(ei, E, N, dinv, buf1, buf2);
    gcn_bias_relu<<<(N * 32 + 255) / 256, 256, 0, stream>>>(buf2, b1, N * 32);

    // --- layer 2: t2 = h @ W2 ; out = scatter(t2) ; log_softmax(out + b2) ---
    gcn_gemm_wmma<64><<<N / 16, 128, 0, stream>>>(buf2, W2, buf1);
    gcn_zero_f4<<<(N * 16 + 255) / 256, 256, 0, stream>>>((float4*)out, N * 16);
    gcn_scatter<64><<<edgeBlocks, 256, 0, stream>>>(ei, E, N, dinv, buf1, out);
    gcn_bias_logsoftmax<<<(N + 7) / 8, 256, 0, stream>>>(out, b2, N);
}